// GraphSage_27212912787602
// MI455X (gfx1250) — compile-verified
//
#include <hip/hip_runtime.h>

typedef __attribute__((ext_vector_type(2))) float v2f;
typedef __attribute__((ext_vector_type(4))) float v4f;
typedef __attribute__((ext_vector_type(8))) float v8f;

#define DIM       128   // INPUT_DIM == HIDDEN_DIM
#define KSLOTS    16    // neighbors per node (incl. self at slot 0)
#define TILE_M    16    // output rows per workgroup (one WMMA M-tile)
#define A_STRIDE  264   // 256 K-values + 8 floats pad (kills LDS bank conflicts)
#define NTHREADS  256   // 8 waves (wave32)

__global__ __launch_bounds__(NTHREADS)
void graphsage_fused_wmma(const int* __restrict__ adj,
                          const float* __restrict__ emb,
                          const float* __restrict__ W,
                          const float* __restrict__ Bmat,
                          float* __restrict__ out)
{
    __shared__ int   s_idx[TILE_M * KSLOTS];
    __shared__ __align__(16) float As[TILE_M * A_STRIDE];

    const int tile = blockIdx.x;
    const int tid  = threadIdx.x;

    // ---- 1) stage adjacency indices for this 16-row tile (256 of them) ----
    s_idx[tid] = adj[(long)tile * (TILE_M * KSLOTS) + tid];
    __syncthreads();

    // ---- 2) gather + neighbor mean -> combined A panel [agg(0..127) | self(128..255)] ----
    const int r = tid >> 4;          // row 0..15
    const int c = (tid & 15) << 3;   // 8 dims per thread
    v4f self0 = {0.f,0.f,0.f,0.f}, self1 = {0.f,0.f,0.f,0.f};
    v4f acc0  = {0.f,0.f,0.f,0.f}, acc1  = {0.f,0.f,0.f,0.f};

    #pragma unroll 4
    for (int slot = 0; slot < KSLOTS; ++slot) {
        const long idx = (long)s_idx[r * KSLOTS + slot];
        const float* p = emb + idx * DIM + c;
        v4f f0 = *(const v4f*)(p);
        v4f f1 = *(const v4f*)(p + 4);
        if (slot == 0) { self0 = f0; self1 = f1; }
        else           { acc0 += f0; acc1 += f1; }
    }
    const float inv = 1.0f / 15.0f;
    float* arow = &As[r * A_STRIDE];
    *(v4f*)(arow + c)           = acc0 * inv;   // agg   -> K = c..c+7
    *(v4f*)(arow + c + 4)       = acc1 * inv;
    *(v4f*)(arow + 128 + c)     = self0;        // self  -> K = 128+c..
    *(v4f*)(arow + 128 + c + 4) = self1;
    __syncthreads();

    // ---- 3) WMMA: wave w computes output columns [16w, 16w+16) over K=256 ----
    const int wave = tid >> 5;
    const int lane = tid & 31;
    const int hi   = lane >> 4;            // 0 -> K pair {0,1}; 1 -> K pair {2,3}
    const int mrow = lane & 15;            // A row for this lane
    const int col  = (wave << 4) + mrow;   // B/N column for this lane

    v8f cacc = {0.f,0.f,0.f,0.f,0.f,0.f,0.f,0.f};
    const float* arow2 = &As[mrow * A_STRIDE];

    // first 128 K-values: agg x W
    #pragma unroll 8
    for (int kk = 0; kk < 32; ++kk) {
        const int kb = (kk << 2) + (hi << 1);
        v2f a = *(const v2f*)(arow2 + kb);                 // ds_load_b64
        v2f bv;
        bv.x = W[(long)kb * DIM + col];
        bv.y = W[(long)(kb + 1) * DIM + col];
        cacc = __builtin_amdgcn_wmma_f32_16x16x4_f32(
                   false, a, false, bv, (short)0, cacc, false, false);
    }
    // last 128 K-values: self x b
    #pragma unroll 8
    for (int kk = 0; kk < 32; ++kk) {
        const int kb = (kk << 2) + (hi << 1);
        v2f a = *(const v2f*)(arow2 + 128 + kb);
        v2f bv;
        bv.x = Bmat[(long)kb * DIM + col];
        bv.y = Bmat[(long)(kb + 1) * DIM + col];
        cacc = __builtin_amdgcn_wmma_f32_16x16x4_f32(
                   false, a, false, bv, (short)0, cacc, false, false);
    }

    // ---- 4) ReLU + store (C layout: VGPR j -> M=j (lanes 0-15) / M=j+8 (lanes 16-31)) ----
    float* orow = out + (long)tile * TILE_M * DIM + col;
    #pragma unroll
    for (int j = 0; j < 8; ++j) {
        const int m = j + (hi << 3);
        float v = cacc[j];
        orow[(long)m * DIM] = v > 0.f ? v : 0.f;
    }
}

extern "C" void kernel_launch(void* const* d_in, const int* in_sizes, int n_in,
                              void* d_out, int out_size, void* d_ws, size_t ws_size,
                              hipStream_t stream)
{
    const int*   adj  = (const int*)d_in[0];    // (16,32,32,16) int32
    const float* emb  = (const float*)d_in[1];  // (1000001,128) f32
    const float* W    = (const float*)d_in[2];  // (128,128) f32
    const float* bmat = (const float*)d_in[3];  // (128,128) f32
    float*       out  = (float*)d_out;          // (16,32,32,128) f32

    const int M_total  = 16 * 32 * 32;          // 16384 output rows
    const int num_tiles = M_total / TILE_M;     // 1024 workgroups

    graphsage_fused_wmma<<<num_tiles, NTHREADS, 0, stream>>>(adj, emb, W, bmat, out);
}